// SVDDLayer_54975581388962
// MI455X (gfx1250) — compile-verified
//
#include <hip/hip_runtime.h>
#include <math.h>

// Problem constants (from reference): inputs (512, 128, 1024) fp32
#define NS   128          // shot (QP dimension)
#define DD   1024         // feature dim
#define NIT  30           // IPM iterations
#define EPSC 1e-6f
#define SIGC 0.1f
#define STR  132          // padded LDS row stride (132 mod 64 = 4 -> conflict-friendly)
#define TPB  256          // 8 waves of 32

typedef float v2f __attribute__((ext_vector_type(2)));
typedef float v8f __attribute__((ext_vector_type(8)));

__device__ __forceinline__ float blk_sum(float v, float* red, int tid) {
    red[tid] = v; __syncthreads();
    #pragma unroll
    for (int s = TPB / 2; s > 0; s >>= 1) {
        if (tid < s) red[tid] += red[tid + s];
        __syncthreads();
    }
    float r = red[0]; __syncthreads();
    return r;
}

__device__ __forceinline__ float blk_min(float v, float* red, int tid) {
    red[tid] = v; __syncthreads();
    #pragma unroll
    for (int s = TPB / 2; s > 0; s >>= 1) {
        if (tid < s) red[tid] = fminf(red[tid], red[tid + s]);
        __syncthreads();
    }
    float r = red[0]; __syncthreads();
    return r;
}

__launch_bounds__(TPB)
__global__ void svdd_center_kernel(const float* __restrict__ X,
                                   float* __restrict__ out) {
    const int b    = blockIdx.x;
    const float* Xb = X + (size_t)b * NS * DD;
    const int tid  = threadIdx.x;
    const int wave = tid >> 5;
    const int lane = tid & 31;
    const int half = lane >> 4;     // 0: lanes 0-15, 1: lanes 16-31
    const int l16  = lane & 15;

    __shared__ float Q[NS * STR];   // Q = 2*(X X^T + eps I)
    __shared__ float M[NS * STR];   // Cholesky workspace
    __shared__ float av[NS], lv[NS], rcv[NS], y1[NS], y2[NS];
    __shared__ float red[TPB];
    __shared__ float nu_s[1];

    // ================= Gram matrix via fp32 WMMA (16x16x4) =================
    // Each wave owns one 16x128 strip of the output: ti = wave, tj = 0..7.
    // Per k-step: 1 A-fragment load shared across 8 WMMAs + 8 B-fragment loads
    // (single base pointer, immediate offsets tj*16*DD*4 = tj*64KB).
    {
        const int ti = wave;
        // A frag: row = ti*16 + l16 ; VGPR0/1 hold K = {0,1} (half 0) or {2,3} (half 1)
        const float* Arow  = Xb + (size_t)(ti * 16 + l16) * DD + 2 * half;
        // B frag base: B[k][n] = X[n][k] -> row n = tj*16 + l16
        const float* Bbase = Xb + (size_t)l16 * DD + 2 * half;

        v8f acc[8];
        #pragma unroll
        for (int tj = 0; tj < 8; ++tj) acc[tj] = (v8f){};

        for (int k = 0; k < DD; k += 4) {
            v2f a = *reinterpret_cast<const v2f*>(Arow + k);    // global_load_b64
            #pragma unroll
            for (int tj = 0; tj < 8; ++tj) {
                v2f bb = *reinterpret_cast<const v2f*>(Bbase + (size_t)tj * 16 * DD + k);
                acc[tj] = __builtin_amdgcn_wmma_f32_16x16x4_f32(
                              false, a, false, bb, (short)0, acc[tj], false, false);
            }
        }

        // C/D layout: VGPR v -> row m = 8*half + v, col n = l16
        #pragma unroll
        for (int tj = 0; tj < 8; ++tj) {
            #pragma unroll
            for (int v = 0; v < 8; ++v) {
                int gr = ti * 16 + 8 * half + v;
                int gc = tj * 16 + l16;
                float q = 2.0f * acc[tj][v];
                if (gr == gc) q += 2.0f * EPSC;
                Q[gr * STR + gc] = q;
            }
        }
    }

    // ================= IPM init =================
    if (tid < NS) { av[tid] = 1.0f / NS; lv[tid] = 1.0f; }
    if (tid == 0) nu_s[0] = 0.0f;
    __syncthreads();

    for (int it = 0; it < NIT; ++it) {
        // r_pri, mu
        float pa  = (tid < NS) ? av[tid] : 0.0f;
        float pal = (tid < NS) ? av[tid] * lv[tid] : 0.0f;
        float r_pri = blk_sum(pa,  red, tid) - 1.0f;
        float mu    = blk_sum(pal, red, tid) / (float)NS;

        // r_dual, r_cent, rhs for solve:  M [v1 v2] = [-(r_dual + r_cent/a), 1]
        if (tid < NS) {
            const float* Qr = &Q[tid * STR];
            float s = 0.0f;
            #pragma unroll 4
            for (int c2 = 0; c2 < NS; ++c2) s += Qr[c2] * av[c2];
            float p  = -0.5f * Q[tid * STR + tid];      // p = -diag(K)
            float rd = s + p - lv[tid] + nu_s[0];
            float rc = av[tid] * lv[tid] - SIGC * mu;
            rcv[tid] = rc;
            y1[tid]  = -(rd + rc / av[tid]);
            y2[tid]  = 1.0f;
        }
        __syncthreads();

        // M = Q + diag(lam/a)
        for (int idx = tid; idx < NS * NS; idx += TPB) {
            int r  = idx >> 7;
            int c2 = idx & (NS - 1);
            float m = Q[r * STR + c2];
            if (r == c2) m += lv[r] / av[r];
            M[r * STR + c2] = m;
        }
        __syncthreads();

        // Cholesky (lower, in place)
        for (int k = 0; k < NS; ++k) {
            if (tid == 0) M[k * STR + k] = sqrtf(M[k * STR + k]);
            __syncthreads();
            float dk = M[k * STR + k];
            for (int i = k + 1 + tid; i < NS; i += TPB) M[i * STR + k] /= dk;
            __syncthreads();
            for (int i = k + 1; i < NS; ++i) {
                float lik = M[i * STR + k];
                for (int j = k + 1 + tid; j <= i; j += TPB)
                    M[i * STR + j] -= lik * M[j * STR + k];
            }
            __syncthreads();
        }

        // Forward solve L y = rhs (both RHS), column-oriented
        for (int j = 0; j < NS; ++j) {
            if (tid == 0) { float d = M[j * STR + j]; y1[j] /= d; y2[j] /= d; }
            __syncthreads();
            float a1 = y1[j], a2 = y2[j];
            for (int i = j + 1 + tid; i < NS; i += TPB) {
                float l = M[i * STR + j];
                y1[i] -= l * a1; y2[i] -= l * a2;
            }
            __syncthreads();
        }
        // Backward solve L^T x = y
        for (int j = NS - 1; j >= 0; --j) {
            if (tid == 0) { float d = M[j * STR + j]; y1[j] /= d; y2[j] /= d; }
            __syncthreads();
            float a1 = y1[j], a2 = y2[j];
            for (int i = tid; i < j; i += TPB) {
                float l = M[j * STR + i];
                y1[i] -= l * a1; y2[i] -= l * a2;
            }
            __syncthreads();
        }

        // dnu, da, dlam, fraction-to-boundary step
        float s1 = (tid < NS) ? y1[tid] : 0.0f;
        float s2 = (tid < NS) ? y2[tid] : 0.0f;
        float S1 = blk_sum(s1, red, tid);
        float S2 = blk_sum(s2, red, tid);
        float dnu = (S1 + r_pri) / S2;

        float rmin = 1e30f, da_ = 0.0f, dl_ = 0.0f;
        if (tid < NS) {
            da_ = y1[tid] - dnu * y2[tid];
            dl_ = (-rcv[tid] - lv[tid] * da_) / av[tid];
            float r1 = (da_ < 0.0f) ? -av[tid] / da_ : 1e30f;
            float r2 = (dl_ < 0.0f) ? -lv[tid] / dl_ : 1e30f;
            rmin = fminf(r1, r2);
        }
        float tstep = fminf(1.0f, 0.99f * blk_min(rmin, red, tid));

        if (tid < NS) { av[tid] += tstep * da_; lv[tid] += tstep * dl_; }
        if (tid == 0) nu_s[0] += tstep * dnu;
        __syncthreads();
    }

    // ================= centers = sum_s a_s * X[s,:] =================
    for (int d = tid; d < DD; d += TPB) {
        float acc = 0.0f;
        #pragma unroll 4
        for (int s = 0; s < NS; ++s) acc += av[s] * Xb[(size_t)s * DD + d];
        out[(size_t)b * DD + d] = acc;
    }
}

extern "C" void kernel_launch(void* const* d_in, const int* in_sizes, int n_in,
                              void* d_out, int out_size, void* d_ws, size_t ws_size,
                              hipStream_t stream) {
    const float* X = (const float*)d_in[0];
    float* out = (float*)d_out;
    const int B = in_sizes[0] / (NS * DD);   // 512
    svdd_center_kernel<<<dim3(B), dim3(TPB), 0, stream>>>(X, out);
}